// EnhancedGNNEncoder_70368744177964
// MI455X (gfx1250) — compile-verified
//
#include <hip/hip_runtime.h>
#include <hip/hip_bf16.h>

#define N_NODES 100000
#define N_EDGES 3200000
#define DIM 64

typedef __attribute__((ext_vector_type(2))) float v2f;
typedef __attribute__((ext_vector_type(8))) float v8f;

// ---------------- zero workspace (graph-capture-safe, no memset) -------------
__global__ __launch_bounds__(256) void zero_kernel(float4* __restrict__ p, long long n4) {
    long long i = (long long)blockIdx.x * blockDim.x + threadIdx.x;
    long long stride = (long long)gridDim.x * blockDim.x;
    for (; i < n4; i += stride) p[i] = make_float4(0.f, 0.f, 0.f, 0.f);
}

// ---------------- per-edge scalar weight: softplus(edge_attr . w + b) --------
__global__ __launch_bounds__(256) void edge_w_kernel(const float* __restrict__ ea,
                                                     const float* __restrict__ mw,
                                                     const float* __restrict__ mb,
                                                     float* __restrict__ w, int E) {
    int e = blockIdx.x * blockDim.x + threadIdx.x;
    if (e >= E) return;
    const float* a = ea + (long long)e * 8;
    float acc = mb[0];
#pragma unroll
    for (int i = 0; i < 8; ++i) acc += a[i] * mw[i];
    // numerically stable softplus: max(x,0) + log1p(exp(-|x|))
    float sp = fmaxf(acc, 0.f) + log1pf(expf(-fabsf(acc)));
    w[e] = sp;
}

// ---------------- edge message + scatter add: agg[dst] += w*(h[src]-h[dst]) --
__global__ __launch_bounds__(256) void edge_scatter_kernel(const float* __restrict__ h,
                                                           const long long* __restrict__ ei,
                                                           const float* __restrict__ w,
                                                           float* __restrict__ agg, int E) {
    long long gid = (long long)blockIdx.x * blockDim.x + threadIdx.x;
    long long total = (long long)E * 16;  // 16 threads per edge, float4 each
    if (gid >= total) return;
    int e  = (int)(gid >> 4);
    int c  = (int)(gid & 15) * 4;
    long long s = ei[e];
    long long d = ei[(long long)E + e];
    float we = w[e];
    float4 a = *(const float4*)(h + s * DIM + c);
    float4 b = *(const float4*)(h + d * DIM + c);
    float* dst = agg + d * DIM + c;
    unsafeAtomicAdd(dst + 0, we * (a.x - b.x));
    unsafeAtomicAdd(dst + 1, we * (a.y - b.y));
    unsafeAtomicAdd(dst + 2, we * (a.z - b.z));
    unsafeAtomicAdd(dst + 3, we * (a.w - b.w));
}

// ---------------- out[n][j] = sum_k A[n][k]*W[j][k] + bias[j] via fp32 WMMA --
// One wave per (16-node block, 16-output-channel tile). K=64 -> 16 steps of
// V_WMMA_F32_16X16X4_F32. Layouts per CDNA5 ISA 7.12.2:
//   A 16x4 : lane(l<16) m=l holds {K=k0, K=k0+1}; lanes 16-31 hold {K=k0+2, k0+3}
//   B 4x16 : vgpr0 = rows k0 (lanes 0-15) / k0+2 (16-31); vgpr1 = k0+1 / k0+3
//   C 16x16: vgpr v = row (v + 8*half), col = lane&15
__global__ __launch_bounds__(256) void gemm64_wmma(const float* __restrict__ A,
                                                   const float* __restrict__ W,
                                                   const float* __restrict__ bias,
                                                   float* __restrict__ out, int nBlocks) {
    int wave = (int)((blockIdx.x * blockDim.x + threadIdx.x) >> 5);
    int lane = threadIdx.x & 31;
    int ntile = wave & 3;
    int nb    = wave >> 2;
    if (nb >= nBlocks) return;

    int half = lane >> 4;        // 0 or 1
    int l15  = lane & 15;
    int j    = ntile * 16 + l15; // output channel owned by this lane (B/C frag)

    // Preload the 64x16 weight tile (transposed view) into 16 v2f fragments.
    const float* Wrow = W + (long long)j * DIM + 2 * half;
    v2f bfrag[16];
#pragma unroll
    for (int s = 0; s < 16; ++s)
        bfrag[s] = *(const v2f*)(Wrow + 4 * s);

    long long node0 = (long long)nb * 16;
    const float* Arow = A + (node0 + l15) * DIM + 2 * half;

    v8f c = {};
#pragma unroll
    for (int s = 0; s < 16; ++s) {
        v2f a = *(const v2f*)(Arow + 4 * s);
        c = __builtin_amdgcn_wmma_f32_16x16x4_f32(
                /*neg_a=*/false, a, /*neg_b=*/false, bfrag[s],
                /*c_mod=*/(short)0, c, /*reuse_a=*/false, /*reuse_b=*/false);
    }

    float bv = bias[j];
    float* orow = out + node0 * DIM + j;
#pragma unroll
    for (int v = 0; v < 8; ++v) {
        int node_off = v + 8 * half;
        orow[(long long)node_off * DIM] = c[v] + bv;
    }
}

// ---------------- ReLU + LayerNorm + residual, one wave per node -------------
__global__ __launch_bounds__(256) void node_finish(const float* __restrict__ g,
                                                   const float* __restrict__ ident,
                                                   const float* __restrict__ gamma,
                                                   const float* __restrict__ beta,
                                                   float* __restrict__ hout, int N) {
    int node = (int)((blockIdx.x * blockDim.x + threadIdx.x) >> 5);
    int lane = threadIdx.x & 31;
    if (node >= N) return;
    const float* row = g + (long long)node * DIM;
    float v0 = fmaxf(row[lane], 0.f);
    float v1 = fmaxf(row[lane + 32], 0.f);
    float sum = v0 + v1;
    float sq  = v0 * v0 + v1 * v1;
#pragma unroll
    for (int off = 16; off > 0; off >>= 1) {
        sum += __shfl_xor(sum, off, 32);
        sq  += __shfl_xor(sq,  off, 32);
    }
    float mean = sum * (1.f / 64.f);
    float var  = sq * (1.f / 64.f) - mean * mean;
    float rstd = rsqrtf(var + 1e-5f);
    const float* idr = ident + (long long)node * DIM;
    float o0 = (v0 - mean) * rstd * gamma[lane]      + beta[lane]      + idr[lane];
    float o1 = (v1 - mean) * rstd * gamma[lane + 32] + beta[lane + 32] + idr[lane + 32];
    float* orow = hout + (long long)node * DIM;
    orow[lane]      = o0;
    orow[lane + 32] = o1;
}

extern "C" void kernel_launch(void* const* d_in, const int* in_sizes, int n_in,
                              void* d_out, int out_size, void* d_ws, size_t ws_size,
                              hipStream_t stream) {
    const float*     x      = (const float*)d_in[0];
    const long long* ei     = (const long long*)d_in[1];
    const float*     ea     = (const float*)d_in[2];
    const float*     lin_w  = (const float*)d_in[3];
    const float*     lin_b  = (const float*)d_in[4];
    const float*     emlp_w = (const float*)d_in[5];
    const float*     emlp_b = (const float*)d_in[6];
    const float*     gamma  = (const float*)d_in[7];
    const float*     beta   = (const float*)d_in[8];
    const float*     fc_w   = (const float*)d_in[9];
    const float*     fc_b   = (const float*)d_in[10];
    float* out = (float*)d_out;

    const int N = N_NODES, E = N_EDGES;
    const size_t ND = (size_t)N * DIM;

    float* agg = (float*)d_ws;
    float* hA  = agg + ND;
    float* hB  = hA + ND;
    float* wgt = hB + ND;     // E floats

    const int nodeBlocks = N / 16;                    // 6250 (exact)
    const int gemmGrid   = (nodeBlocks * 4 + 7) / 8;  // 8 waves / 256-thread block
    const long long scatterTot = (long long)E * 16;

    const float* hcur = x;
    float* hbufs[2] = {hA, hB};
    for (int l = 0; l < 3; ++l) {
        edge_w_kernel<<<(E + 255) / 256, 256, 0, stream>>>(
            ea, emlp_w + l * 8, emlp_b + l, wgt, E);
        zero_kernel<<<2048, 256, 0, stream>>>((float4*)agg, (long long)(ND / 4));
        edge_scatter_kernel<<<(unsigned)((scatterTot + 255) / 256), 256, 0, stream>>>(
            hcur, ei, wgt, agg, E);
        // GEMM result goes into d_out (reused as scratch; fully overwritten)
        gemm64_wmma<<<gemmGrid, 256, 0, stream>>>(
            agg, lin_w + (size_t)l * DIM * DIM, lin_b + l * DIM, out, nodeBlocks);
        node_finish<<<(N + 7) / 8, 256, 0, stream>>>(
            out, hcur, gamma + l * DIM, beta + l * DIM, hbufs[l & 1], N);
        hcur = hbufs[l & 1];
    }
    // Final projection: h @ fc_w.T + fc_b -> d_out
    gemm64_wmma<<<gemmGrid, 256, 0, stream>>>(hcur, fc_w, fc_b, out, nodeBlocks);
}